// DeepAR_88484916232673
// MI455X (gfx1250) — compile-verified
//
#include <hip/hip_runtime.h>
#include <math.h>

// ---------------- problem constants ----------------
namespace {
constexpr int kN    = 4096;
constexpr int kSeq  = 168;
constexpr int kHor  = 24;
constexpr int kT    = kSeq + kHor;   // 192
constexpr int kHid  = 64;            // LSTM in/out dim (= F_IN + EMBED)
constexpr int kHStr = 68;            // padded LDS row stride (floats) -> no bank conflicts
constexpr size_t kMuOff  = (size_t)kN * kHor;
constexpr size_t kSigOff = kMuOff + (size_t)kN * kT;
}

typedef __attribute__((ext_vector_type(2))) float v2f;   // f32 WMMA A/B fragment (16x4 / 4x16)
typedef __attribute__((ext_vector_type(8))) float v8f;   // f32 WMMA C/D fragment (16x16)

__device__ __forceinline__ float sigmoidf_(float x) { return 1.0f / (1.0f + expf(-x)); }

// One LSTM "cell from zero state" layer on 16 rows held in a per-wave LDS buffer
// hb: 16 rows x 64 cols (stride kHStr). gates = h @ W^T + b_ih + b_hh, torch order i,f,g,o.
// f-gate is skipped (c0 == 0). Output h overwrites hb.
__device__ __forceinline__ void lstm_layer_wmma(float* hb, const float* __restrict__ W,
                                                const float* __restrict__ bih,
                                                const float* __restrict__ bhh,
                                                bool do_relu, int row, int half)
{
    // A fragments: 16x4 f32 tile kb -> lane(M=row,halfK) holds contiguous float2 of h row
    v2f a[16];
#pragma unroll
    for (int kb = 0; kb < 16; ++kb)
        a[kb] = *(const v2f*)(hb + row * kHStr + 4 * kb + 2 * half);
    __syncthreads();   // everyone done reading h before we overwrite it below

#pragma unroll
    for (int nt = 0; nt < 4; ++nt) {
        const int gi = nt * 16 + row;     // i-gate output index for this lane's N column
        const int gg = gi + 128;          // g-gate
        const int go = gi + 192;          // o-gate
        float bi = bih[gi] + bhh[gi];
        float bg = bih[gg] + bhh[gg];
        float bo = bih[go] + bhh[go];
        v8f ci, cg, co;
#pragma unroll
        for (int j = 0; j < 8; ++j) { ci[j] = bi; cg[j] = bg; co[j] = bo; }

        // B fragments: 4x16 tile -> lane(N=row,halfK) = contiguous float2 of W row
        const float* wi = W + (size_t)gi * kHid + 2 * half;
        const float* wg = W + (size_t)gg * kHid + 2 * half;
        const float* wo = W + (size_t)go * kHid + 2 * half;
#pragma unroll
        for (int kb = 0; kb < 16; ++kb) {
            v2f fbi = *(const v2f*)(wi + 4 * kb);
            v2f fbg = *(const v2f*)(wg + 4 * kb);
            v2f fbo = *(const v2f*)(wo + 4 * kb);
            ci = __builtin_amdgcn_wmma_f32_16x16x4_f32(false, a[kb], false, fbi, (short)0, ci, false, false);
            cg = __builtin_amdgcn_wmma_f32_16x16x4_f32(false, a[kb], false, fbg, (short)0, cg, false, false);
            co = __builtin_amdgcn_wmma_f32_16x16x4_f32(false, a[kb], false, fbo, (short)0, co, false, false);
        }
        // c = sig(i)*tanh(g); h = sig(o)*tanh(c); write back in C-tile layout (M=j+8*half, N=row)
#pragma unroll
        for (int j = 0; j < 8; ++j) {
            float c = sigmoidf_(ci[j]) * tanhf(cg[j]);
            float h = sigmoidf_(co[j]) * tanhf(c);
            if (do_relu) h = fmaxf(h, 0.0f);
            hb[(j + 8 * half) * kHStr + nt * 16 + row] = h;
        }
    }
    __syncthreads();
}

__device__ __forceinline__ void head_eval(const float* hb, int row, int half,
                                          const float* __restrict__ Wmu, const float* __restrict__ bmu,
                                          const float* __restrict__ Wsg, const float* __restrict__ bsg,
                                          float ynext, float& mu, float& sigma, float& lik)
{
    float pm = 0.0f, ps = 0.0f;
#pragma unroll
    for (int k = 0; k < 32; ++k) {
        float hv = hb[row * kHStr + half * 32 + k];
        pm = fmaf(hv, Wmu[half * 32 + k], pm);
        ps = fmaf(hv, Wsg[half * 32 + k], ps);
    }
    pm += __shfl_xor(pm, 16, 32);
    ps += __shfl_xor(ps, 16, 32);
    mu = pm + bmu[0];
    float sp = ps + bsg[0];
    sigma = fmaxf(sp, 0.0f) + log1pf(expf(-fabsf(sp))) + 1e-6f;  // jax softplus + eps
    float d  = ynext - mu;
    float s2 = sigma * sigma;
    lik = expf(-(d * d) / (2.0f * s2)) / sqrtf(6.283185307179586f * s2);
}

// ---------------- Phase 1: teacher-forced steps t=0..167, all (n,t) independent ----------------
__global__ __launch_bounds__(256) void deepar_tf_kernel(
    const float* __restrict__ X, const float* __restrict__ y,
    const float* __restrict__ W_embed, const float* __restrict__ b_embed,
    const float* __restrict__ W_ih, const float* __restrict__ b_ih, const float* __restrict__ b_hh,
    const float* __restrict__ W_mu, const float* __restrict__ b_mu,
    const float* __restrict__ W_sig, const float* __restrict__ b_sig,
    float* __restrict__ out)
{
    __shared__ float lds[8 * 16 * kHStr];
    const int tid  = threadIdx.x;
    const int wave = tid >> 5;
    const int lane = tid & 31;
    const int row  = lane & 15, half = lane >> 4;
    float* hb = lds + wave * 16 * kHStr;
    const size_t r0 = ((size_t)blockIdx.x * 8 + wave) * 16;   // flat row r = n*168 + t

    // h0 cols 0..31 = X rows (contiguous since X is (N,168,32) and r is the flat (n,t) index)
#pragma unroll
    for (int m = 0; m < 16; ++m)
        hb[m * kHStr + lane] = X[r0 * 32 + (size_t)m * 32 + lane];
    // h0 cols 32..63 = y * W_embed + b_embed (lane == embed index)
    const float we = W_embed[lane], be = b_embed[lane];
#pragma unroll
    for (int m = 0; m < 16; ++m)
        hb[m * kHStr + 32 + lane] = fmaf(y[r0 + m], we, be);
    __syncthreads();

    lstm_layer_wmma(hb, W_ih,            b_ih,       b_hh,       false, row, half);
    lstm_layer_wmma(hb, W_ih + 256 * 64, b_ih + 256, b_hh + 256, true,  row, half);

    float mu, sigma, lik;
    const float ynext = y[r0 + row];
    head_eval(hb, row, half, W_mu, b_mu, W_sig, b_sig, ynext, mu, sigma, lik);

    const size_t r = r0 + row;
    const int n = (int)(r / kSeq);
    const int t = (int)(r % kSeq);
    if (half == 0) {
        out[kMuOff  + (size_t)n * kT + t] = mu;
        out[kSigOff + (size_t)n * kT + t] = sigma;
        if (t == kSeq - 1)
            out[(size_t)n * kHor + 0] = lik;   // ypred col 0 = lik[167]
    }
}

// ---------------- Phase 2: autoregressive step t (168..191), carry = ypred col t-168 ----------------
__global__ __launch_bounds__(256) void deepar_ar_kernel(
    const float* __restrict__ Xf,
    const float* __restrict__ W_embed, const float* __restrict__ b_embed,
    const float* __restrict__ W_ih, const float* __restrict__ b_ih, const float* __restrict__ b_hh,
    const float* __restrict__ W_mu, const float* __restrict__ b_mu,
    const float* __restrict__ W_sig, const float* __restrict__ b_sig,
    float* __restrict__ out, int t)
{
    __shared__ float lds[8 * 16 * kHStr];
    const int tf   = t - kSeq;           // 0..23
    const int tid  = threadIdx.x;
    const int wave = tid >> 5;
    const int lane = tid & 31;
    const int row  = lane & 15, half = lane >> 4;
    float* hb = lds + wave * 16 * kHStr;
    const int n0 = (blockIdx.x * 8 + wave) * 16;
    float* ypred = out;

#pragma unroll
    for (int m = 0; m < 16; ++m)
        hb[m * kHStr + lane] = Xf[((size_t)(n0 + m) * kHor + tf) * 32 + lane];
    const float we = W_embed[lane], be = b_embed[lane];
#pragma unroll
    for (int m = 0; m < 16; ++m) {
        float yv = ypred[(size_t)(n0 + m) * kHor + tf];   // carry: lik[t-1]
        hb[m * kHStr + 32 + lane] = fmaf(yv, we, be);
    }
    __syncthreads();

    lstm_layer_wmma(hb, W_ih,            b_ih,       b_hh,       false, row, half);
    lstm_layer_wmma(hb, W_ih + 256 * 64, b_ih + 256, b_hh + 256, true,  row, half);

    float mu, sigma, lik;
    const float ynext = ypred[(size_t)(n0 + row) * kHor + tf];
    head_eval(hb, row, half, W_mu, b_mu, W_sig, b_sig, ynext, mu, sigma, lik);

    if (half == 0) {
        out[kMuOff  + (size_t)(n0 + row) * kT + t] = mu;
        out[kSigOff + (size_t)(n0 + row) * kT + t] = sigma;
        if (tf < kHor - 1)
            ypred[(size_t)(n0 + row) * kHor + tf + 1] = lik;  // lik[t] -> ypred col t-167
    }
}

extern "C" void kernel_launch(void* const* d_in, const int* in_sizes, int n_in,
                              void* d_out, int out_size, void* d_ws, size_t ws_size,
                              hipStream_t stream)
{
    (void)in_sizes; (void)n_in; (void)out_size; (void)d_ws; (void)ws_size;
    const float* X       = (const float*)d_in[0];
    const float* y       = (const float*)d_in[1];
    const float* Xf      = (const float*)d_in[2];
    const float* W_embed = (const float*)d_in[3];
    const float* b_embed = (const float*)d_in[4];
    const float* W_ih    = (const float*)d_in[5];
    const float* b_ih    = (const float*)d_in[6];
    const float* b_hh    = (const float*)d_in[7];
    const float* W_mu    = (const float*)d_in[8];
    const float* b_mu    = (const float*)d_in[9];
    const float* W_sig   = (const float*)d_in[10];
    const float* b_sig   = (const float*)d_in[11];
    float* out = (float*)d_out;

    dim3 blk(256);
    // Phase 1: 4096*168 rows, 128 rows per block -> 5376 blocks
    dim3 g1((kN * kSeq) / 128);
    deepar_tf_kernel<<<g1, blk, 0, stream>>>(X, y, W_embed, b_embed, W_ih, b_ih, b_hh,
                                             W_mu, b_mu, W_sig, b_sig, out);
    // Phase 2: 24 sequential autoregressive steps, 4096 rows each
    dim3 g2(kN / 128);
    for (int t = kSeq; t < kT; ++t)
        deepar_ar_kernel<<<g2, blk, 0, stream>>>(Xf, W_embed, b_embed, W_ih, b_ih, b_hh,
                                                 W_mu, b_mu, W_sig, b_sig, out, t);
}